// CustomPoseLoss_44169443672307
// MI455X (gfx1250) — compile-verified
//
#include <hip/hip_runtime.h>
#include <hip/hip_bf16.h>

typedef __attribute__((ext_vector_type(2))) float v2f;
typedef __attribute__((ext_vector_type(8))) float v8f;

// ---------------------------------------------------------------------------
// Wave-wide sum using V_WMMA_F32_16X16X4_F32.
// A (16x4): vgpr0 = per-lane value (K0 lanes0-15 / K2 lanes16-31), vgpr1 = 0.
// B (4x16): all ones.  D[m][n] = v[m] + v[m+16]  (broadcast over n).
// Each lane sums its 8 accumulator vgprs -> lanes 0-15 hold sum(v[0..7],v[16..23]),
// lanes 16-31 hold sum(v[8..15],v[24..31]); shfl_xor(16) yields the full total
// in every lane.  EXEC must be all-ones at the call site (it is: uniform flow).
// ---------------------------------------------------------------------------
__device__ __forceinline__ float wave_sum32(float v) {
  v2f a; a.x = v;    a.y = 0.0f;
  v2f b; b.x = 1.0f; b.y = 1.0f;
  v8f c = {};
  c = __builtin_amdgcn_wmma_f32_16x16x4_f32(false, a, false, b,
                                            (short)0, c, false, false);
  float s = c[0] + c[1] + c[2] + c[3] + c[4] + c[5] + c[6] + c[7];
  s += __shfl_xor(s, 16, 32);
  return s;
}

// ---------------------------------------------------------------------------
// Orthogonal polar factor of a 3x3 via determinantally-scaled Newton:
//   X <- 0.5 * ( g*X + cof(X) / (g*det) ),  g = |det|^(-1/3)
// Converges to Q = U*Vt of the SVD (det(Q) = sign(det(M))).  Then apply the
// reference's sign fix: multiply the whole matrix by sign(det).
// ---------------------------------------------------------------------------
#define POLAR_SCALED_ITERS 3
#define POLAR_PLAIN_ITERS  2

__device__ __forceinline__ void polar3x3(const float m[9], float q[9]) {
  float x[9];
#pragma unroll
  for (int i = 0; i < 9; ++i) x[i] = m[i];
  float det0 = 0.0f;
#pragma unroll
  for (int it = 0; it < POLAR_SCALED_ITERS + POLAR_PLAIN_ITERS; ++it) {
    float c0 = x[4] * x[8] - x[5] * x[7];
    float c1 = x[5] * x[6] - x[3] * x[8];
    float c2 = x[3] * x[7] - x[4] * x[6];
    float c3 = x[2] * x[7] - x[1] * x[8];
    float c4 = x[0] * x[8] - x[2] * x[6];
    float c5 = x[1] * x[6] - x[0] * x[7];
    float c6 = x[1] * x[5] - x[2] * x[4];
    float c7 = x[2] * x[3] - x[0] * x[5];
    float c8 = x[0] * x[4] - x[1] * x[3];
    float det = x[0] * c0 + x[1] * c1 + x[2] * c2;
    if (it == 0) det0 = det;
    float ad = fmaxf(fabsf(det), 1e-30f);
    float g = 1.0f;
    if (it < POLAR_SCALED_ITERS)
      g = exp2f(-0.3333333333f * log2f(ad));   // |det|^(-1/3)
    float ha = 0.5f * g;
    float hb = 0.5f / (g * det);               // signed det -> X^{-T} = cof/det
    x[0] = ha * x[0] + hb * c0;
    x[1] = ha * x[1] + hb * c1;
    x[2] = ha * x[2] + hb * c2;
    x[3] = ha * x[3] + hb * c3;
    x[4] = ha * x[4] + hb * c4;
    x[5] = ha * x[5] + hb * c5;
    x[6] = ha * x[6] + hb * c6;
    x[7] = ha * x[7] + hb * c7;
    x[8] = ha * x[8] + hb * c8;
  }
  float s = (det0 >= 0.0f) ? 1.0f : -1.0f;
#pragma unroll
  for (int i = 0; i < 9; ++i) q[i] = s * x[i];
}

// ---------------------------------------------------------------------------
// Main kernel: grid-stride over matrices; per-block (mse, rot) partial sums.
// ---------------------------------------------------------------------------
__global__ __launch_bounds__(256) void pose_loss_partial_kernel(
    const float* __restrict__ pred, const float* __restrict__ target,
    float* __restrict__ partial, int nitems) {
  const int tid = threadIdx.x;
  const int stride = gridDim.x * blockDim.x;
  float acc_mse = 0.0f, acc_rot = 0.0f;

  for (int i = blockIdx.x * blockDim.x + tid; i < nitems; i += stride) {
    const float* p = pred + 9 * (size_t)i;
    const float* t = target + 9 * (size_t)i;
    if (i + stride < nitems) {  // hide HBM latency for the next tile
      __builtin_prefetch(pred + 9 * (size_t)(i + stride), 0, 0);
      __builtin_prefetch(target + 9 * (size_t)(i + stride), 0, 0);
    }
    float pm[9], tm[9];
#pragma unroll
    for (int k = 0; k < 9; ++k) { pm[k] = p[k]; tm[k] = t[k]; }
#pragma unroll
    for (int k = 0; k < 9; ++k) {
      float d = pm[k] - tm[k];
      acc_mse = fmaf(d, d, acc_mse);
    }
    float qp[9], qt[9];
    polar3x3(pm, qp);
    polar3x3(tm, qt);
#pragma unroll
    for (int k = 0; k < 9; ++k) {
      float d = qp[k] - qt[k];
      acc_rot = fmaf(d, d, acc_rot);
    }
  }

  // Uniform control flow here -> EXEC all-ones for WMMA.
  float wm = wave_sum32(acc_mse);
  float wr = wave_sum32(acc_rot);

  __shared__ float sm[8];
  __shared__ float sr[8];
  const int wid = tid >> 5, lane = tid & 31;
  if (lane == 0) { sm[wid] = wm; sr[wid] = wr; }
  __syncthreads();
  if (tid == 0) {
    float a = 0.0f, b = 0.0f;
#pragma unroll
    for (int w = 0; w < 8; ++w) { a += sm[w]; b += sr[w]; }
    partial[2 * blockIdx.x + 0] = a;
    partial[2 * blockIdx.x + 1] = b;
  }
}

// ---------------------------------------------------------------------------
// Finalizer: one wave deterministically folds the per-block partials.
// ---------------------------------------------------------------------------
__global__ __launch_bounds__(32) void pose_loss_finalize_kernel(
    const float* __restrict__ partial, int nparts,
    float* __restrict__ out, float scale) {
  const int lane = threadIdx.x;  // 32 threads
  float m = 0.0f, r = 0.0f;
  for (int j = lane; j < nparts; j += 32) {
    m += partial[2 * j + 0];
    r += partial[2 * j + 1];
  }
  m = wave_sum32(m);
  r = wave_sum32(r);
  if (lane == 0) out[0] = (m + 0.5f * r) * scale;
}

extern "C" void kernel_launch(void* const* d_in, const int* in_sizes, int n_in,
                              void* d_out, int out_size, void* d_ws, size_t ws_size,
                              hipStream_t stream) {
  const float* pred = (const float*)d_in[0];
  const float* target = (const float*)d_in[1];
  float* out = (float*)d_out;
  float* partial = (float*)d_ws;

  const int nitems = in_sizes[0] / 9;  // B = 1,048,576
  int blocks = (nitems + 255) / 256;
  if (blocks > 1024) blocks = 1024;    // 2*blocks floats of scratch (8 KB)

  pose_loss_partial_kernel<<<blocks, 256, 0, stream>>>(pred, target, partial,
                                                       nitems);
  const float scale = 1.0f / (9.0f * (float)nitems);
  pose_loss_finalize_kernel<<<1, 32, 0, stream>>>(partial, blocks, out, scale);
}